// JointMembership_QuantumCircuit_89816356094165
// MI455X (gfx1250) — compile-verified
//
#include <hip/hip_runtime.h>
#include <stdint.h>

// --- CDNA5 (gfx1250) streaming quantum-circuit kernel ---------------------
// Closed form of the reference circuit:
//   z0 = cos(ty)*cos(x0) - sin(ty)*cos(tz)*sin(x0)*sin(x1)
//   z1 = cos(ty)*cos(x1)
//   out = (z + 1) * 0.5
// Memory-bound: ~33.5 MB of traffic -> ~1.4us at 23.3 TB/s.
// Data movement uses the Tensor Data Mover (tensor_load_to_lds) with
// double-buffered LDS tiles and s_wait_tensorcnt for overlap.

typedef unsigned int u32x4 __attribute__((ext_vector_type(4)));
typedef int          i32x4 __attribute__((ext_vector_type(4)));
typedef int          i32x8 __attribute__((ext_vector_type(8)));
typedef float        f32x4 __attribute__((ext_vector_type(4)));
typedef float        f32x2 __attribute__((ext_vector_type(2)));

#define BLOCK_THREADS   256   // 8 wave32 per block
#define TILE_ELEMS      1024  // float2 elements per TDM tile (8 KiB)
#define TILES_PER_BLOCK 4     // double-buffered pipeline depth

// Issue one TDM DMA: copy up to TILE_ELEMS float2 (8-byte units) from global
// memory into LDS at ldsByteOff. tensor_dim0 is clamped to the remaining
// element count so out-of-bounds reads return zero per the TDM spec.
static __device__ __forceinline__ void tdm_load_tile(const float* x,
                                                     long long elemBase,
                                                     unsigned ldsByteOff,
                                                     unsigned remaining8B) {
  unsigned long long g =
      (unsigned long long)(uintptr_t)x + (unsigned long long)elemBase * 8ull;

  // D# group 0: count=1, lds_addr, global_addr[56:0], type=2 ("image")
  u32x4 g0;
  g0.x = 1u;                                      // count = 1 (valid D#)
  g0.y = ldsByteOff;                              // LDS dest byte address
  g0.z = (unsigned)(g & 0xffffffffull);           // global_addr[31:0]
  g0.w = (unsigned)((g >> 32) & 0x01ffffffull)    // global_addr[56:32]
         | (2u << 30);                            // type = 2

  // D# group 1: data_size=3 (8B), 1-D tile of TILE_ELEMS, dims clamped.
  unsigned td0 = remaining8B;                     // tensor_dim0 (OOB clamp)
  i32x8 g1;
  g1[0] = (int)(3u << 16);                        // data_size = 3 -> 8 bytes
  g1[1] = (int)((td0 & 0xffffu) << 16);           // tensor_dim0[15:0]
  g1[2] = (int)((td0 >> 16) | (1u << 16));        // tensor_dim0[31:16], dim1=1
  g1[3] = (int)((unsigned)TILE_ELEMS << 16);      // tile_dim0
  g1[4] = 0;                                      // tile_dim1/2 unused (1-D)
  g1[5] = (int)td0;                               // tensor_dim0_stride
  g1[6] = 0;
  g1[7] = 0;

  i32x4 z4 = {0, 0, 0, 0};
#if defined(__clang_major__) && (__clang_major__ >= 23)
  i32x8 z8 = {0, 0, 0, 0, 0, 0, 0, 0};
  __builtin_amdgcn_tensor_load_to_lds(g0, g1, z4, z4, z8, 0);
#else
  __builtin_amdgcn_tensor_load_to_lds(g0, g1, z4, z4, 0);
#endif
}

__global__ __launch_bounds__(BLOCK_THREADS) void qc_closedform_tdm_kernel(
    const float* __restrict__ x, const float* __restrict__ theta_rz,
    const float* __restrict__ theta_ry, float* __restrict__ out, int nelems) {
  __shared__ f32x4 buf[2][TILE_ELEMS / 2];  // 2 x 8 KiB double buffer

  // Wave-uniform circuit constants (scalar loads + a few trans ops).
  const float ty = theta_ry[0];
  const float tz = theta_rz[0];
  const float A = __cosf(ty);
  const float B = __sinf(ty) * __cosf(tz);

  const int tid = threadIdx.x;
  const bool issuerWave = (tid < 32);  // wave 0 drives the TDM
  const long long blockBase =
      (long long)blockIdx.x * (TILES_PER_BLOCK * TILE_ELEMS);

  // Prime the pipeline: DMA tile 0.
  if (issuerWave) {
    if (blockBase < (long long)nelems)
      tdm_load_tile(x, blockBase, (unsigned)(uintptr_t)&buf[0][0],
                    (unsigned)((long long)nelems - blockBase));
  }

  for (int t = 0; t < TILES_PER_BLOCK; ++t) {
    // Prefetch tile t+1 into the other buffer, then wait for tile t.
    if (t + 1 < TILES_PER_BLOCK) {
      if (issuerWave) {
        const long long eb = blockBase + (long long)(t + 1) * TILE_ELEMS;
        if (eb < (long long)nelems)
          tdm_load_tile(x, eb, (unsigned)(uintptr_t)&buf[(t + 1) & 1][0],
                        (unsigned)((long long)nelems - eb));
      }
      __builtin_amdgcn_s_wait_tensorcnt(1);  // oldest DMA (tile t) complete
    } else {
      __builtin_amdgcn_s_wait_tensorcnt(0);  // drain
    }
    __syncthreads();  // make tile t visible to all 8 waves

    const f32x4* src = buf[t & 1];
    const long long tileBase = blockBase + (long long)t * TILE_ELEMS;

#pragma unroll
    for (int k = 0; k < TILE_ELEMS / (2 * BLOCK_THREADS); ++k) {
      const int vi = k * BLOCK_THREADS + tid;  // float4 index in tile
      const f32x4 v = src[vi];                 // ds_load_b128, conflict-free
      f32x4 r;
      float s0, c0, s1, c1;
      __sincosf(v.x, &s0, &c0);
      __sincosf(v.y, &s1, &c1);
      r.x = fmaf(0.5f, fmaf(-B * s1, s0, A * c0), 0.5f);
      r.y = fmaf(0.5f, A * c1, 0.5f);
      __sincosf(v.z, &s0, &c0);
      __sincosf(v.w, &s1, &c1);
      r.z = fmaf(0.5f, fmaf(-B * s1, s0, A * c0), 0.5f);
      r.w = fmaf(0.5f, A * c1, 0.5f);

      const long long e0 = tileBase + 2LL * vi;  // first float2 element
      if (e0 + 1 < (long long)nelems) {
        // Streaming store: written once, never re-read -> non-temporal b128.
        __builtin_nontemporal_store(r, reinterpret_cast<f32x4*>(out) + (e0 >> 1));
      } else if (e0 < (long long)nelems) {
        f32x2 r2 = {r.x, r.y};
        __builtin_nontemporal_store(r2, reinterpret_cast<f32x2*>(out) + e0);
      }
    }
    __syncthreads();  // all reads done before this buffer is re-filled
  }
}

extern "C" void kernel_launch(void* const* d_in, const int* in_sizes, int n_in,
                              void* d_out, int out_size, void* d_ws,
                              size_t ws_size, hipStream_t stream) {
  (void)n_in; (void)out_size; (void)d_ws; (void)ws_size;
  const float* x  = (const float*)d_in[0];
  const float* tz = (const float*)d_in[1];
  const float* ty = (const float*)d_in[2];
  float* out = (float*)d_out;

  const int nelems = in_sizes[0] / 2;  // batch size (float2 elements)
  const int perBlock = TILE_ELEMS * TILES_PER_BLOCK;
  const int blocks = (nelems + perBlock - 1) / perBlock;  // 512 for BATCH=2^21

  qc_closedform_tdm_kernel<<<blocks, BLOCK_THREADS, 0, stream>>>(x, tz, ty, out,
                                                                 nelems);
}